// FCGATLayer_76192719831645
// MI455X (gfx1250) — compile-verified
//
#include <hip/hip_runtime.h>

// MI455X (gfx1250) fused GAT layer.
// Algebraic folds: e_att = edge @ (W_e folded with att_edge)  [E x H];
//                  agg_edge = (att^T @ edge_row) @ W_m ; softmax rows sum to 1 so b_m passes through.
// All GEMM work uses V_WMMA_F32_16X16X4_F32 (f32 in, f32 acc) -- kernel is HBM-bound (~64MB edge stream).
// A-fragments are loaded as v2f (global_load_b64); edge row blocks are WGP-prefetched
// (global_prefetch_b8) since each block reads its 64KB row twice.

typedef __attribute__((ext_vector_type(2))) float v2f;
typedef __attribute__((ext_vector_type(8))) float v8f;

#define BDIM 4
#define NN 256
#define DD 256
#define EE 64
#define HH 8
#define DH 32
#define LRELU_SLOPE 0.1f
#define NEG_INF (-4294967295.0f)

// ---------------------------------------------------------------------------
// Generic f32 WMMA GEMM: C[M,N] = A[M,K] @ B[K,N]  (+bias, +relu fused)
// One wave per 16x16 C tile, 8 waves per block.
// ---------------------------------------------------------------------------
__global__ __launch_bounds__(256)
void gemm_f32_wmma(const float* __restrict__ A, const float* __restrict__ B,
                   const float* __restrict__ bias, float* __restrict__ C,
                   int M, int N, int K, int fuse /*0=none,1=bias,2=bias+relu*/)
{
    const int lane  = threadIdx.x & 31;
    const int wave  = threadIdx.x >> 5;
    const int tile  = blockIdx.x * 8 + wave;
    const int ntn   = N >> 4;
    const int tm    = (tile / ntn) << 4;
    const int tn    = (tile % ntn) << 4;
    if (tm >= M) return;                         // wave-uniform
    const int m16   = lane & 15;
    const int koff  = (lane >> 4) << 1;          // 0 or 2
    const int rbase = (lane >> 4) << 3;          // 0 or 8

    v8f acc = {0.f,0.f,0.f,0.f,0.f,0.f,0.f,0.f};
    const float* arow = A + (size_t)(tm + m16) * K;       // rows 1KB-aligned
    for (int k0 = 0; k0 < K; k0 += 4) {
        v2f a = *(const v2f*)(arow + k0 + koff);          // global_load_b64
        v2f b;
        b.x = B[(size_t)(k0 + koff)     * N + tn + m16];
        b.y = B[(size_t)(k0 + koff + 1) * N + tn + m16];
        acc = __builtin_amdgcn_wmma_f32_16x16x4_f32(false, a, false, b,
                                                    (short)0, acc, false, false);
    }
#pragma unroll
    for (int r = 0; r < 8; ++r) {
        int row = tm + rbase + r;
        int col = tn + m16;
        float v = acc[r];
        if (fuse >= 1) v += bias[col];
        if (fuse == 2) v = fmaxf(v, 0.f);
        C[(size_t)row * N + col] = v;
    }
}

// ---------------------------------------------------------------------------
// Per-node src/dst attention scores: s/d[b,n,h] = <x_proj[b,n,h,:], att_*[h,:]>
// ---------------------------------------------------------------------------
__global__ void sdatt_kernel(const float* __restrict__ x_proj,
                             const float* __restrict__ att_src,
                             const float* __restrict__ att_dst,
                             float* __restrict__ s_att, float* __restrict__ d_att)
{
    int t = blockIdx.x * blockDim.x + threadIdx.x;   // B*N*H = 8192
    if (t >= BDIM * NN * HH) return;
    int h  = t & 7;
    int bn = t >> 3;
    const float* xp = x_proj + (size_t)bn * DD + h * DH;
    const float* as = att_src + h * DH;
    const float* ad = att_dst + h * DH;
    float s = 0.f, d = 0.f;
#pragma unroll
    for (int k = 0; k < DH; ++k) { s += xp[k] * as[k]; d += xp[k] * ad[k]; }
    s_att[t] = s;
    d_att[t] = d;
}

// ---------------------------------------------------------------------------
// Fold W_e with att_edge:  w_att_e16[e][h16] = sum_d W_e[e, h*Dh+d]*att_edge[h,d]
// (padded to 16 head columns with zeros so it drops straight into WMMA B frags)
// ---------------------------------------------------------------------------
__global__ void fold_kernel(const float* __restrict__ W_e,
                            const float* __restrict__ att_edge,
                            float* __restrict__ w_att_e16)   // [EE][16]
{
    int t = blockIdx.x * blockDim.x + threadIdx.x;   // EE*16 = 1024
    if (t >= EE * 16) return;
    int h = t & 15;
    int e = t >> 4;
    float s = 0.f;
    if (h < HH) {
#pragma unroll
        for (int d = 0; d < DH; ++d)
            s += W_e[(size_t)e * DD + h * DH + d] * att_edge[h * DH + d];
    }
    w_att_e16[t] = s;
}

// ---------------------------------------------------------------------------
// Main attention kernel: one block per (b,i) row.
//   Phase B: e_att via WMMA (edge_row[256x64] @ w_att_e16[64x16]) -> scores in LDS
//   Phase C: per-head softmax (one wave32 per head, shfl_xor reductions)
//   Phase D: T = att^T[16x256] @ edge_row[256x64] via WMMA (waves 0-3)
//   Phase E: agg_x[h,d] = sum_j att[j,h]*x_proj[b,j,h,d]   (waves 4-7, x_proj L2-resident)
//   Phase F: x_out = T @ W_m(head cols) + b_m + agg_x + bias_node
// ---------------------------------------------------------------------------
__global__ __launch_bounds__(256)
void attn_row_kernel(const float* __restrict__ edge_feats,
                     const unsigned char* __restrict__ mask,
                     const float* __restrict__ x_proj,
                     const float* __restrict__ s_att,
                     const float* __restrict__ d_att,
                     const float* __restrict__ w_att_e16,
                     const float* __restrict__ W_m,
                     const float* __restrict__ b_m,
                     const float* __restrict__ bias_node,
                     float* __restrict__ x_out)
{
    __shared__ float lds_att[NN * 16];   // [j][h16]  (h>=8 stays 0)
    __shared__ float lds_T[16 * EE];     // [h16][e]
    __shared__ float lds_aggx[DD];       // [h*32+d]

    const int bi   = blockIdx.x;         // b*256 + i
    const int b    = bi >> 8;
    const int tid  = threadIdx.x;
    const int lane = tid & 31;
    const int wave = tid >> 5;
    const int m16  = lane & 15;
    const int koff = (lane >> 4) << 1;
    const int rbase = (lane >> 4) << 3;

    const float* erow = edge_feats + (size_t)bi * NN * EE;   // [256][64] = 64KB
    const unsigned char* mrow = mask + (size_t)bi * NN;

    // Warm WGP-path for the 64KB edge row block (it is read twice: Phase B + D).
    // 512 prefetches at 128B stride -> global_prefetch_b8 covering the block.
    __builtin_prefetch(erow + (size_t)tid * 64, 0, 3);
    __builtin_prefetch(erow + (size_t)tid * 64 + 32, 0, 3);

    // ---- Phase B: e_att + scores --------------------------------------
    for (int tj = wave; tj < 16; tj += 8) {          // wave-uniform loop
        const int j0 = tj << 4;
        v8f acc = {0.f,0.f,0.f,0.f,0.f,0.f,0.f,0.f};
        const float* arow = erow + (size_t)(j0 + m16) * EE;  // rows 256B-aligned
        for (int k0 = 0; k0 < EE; k0 += 4) {
            v2f a = *(const v2f*)(arow + k0 + koff);         // global_load_b64
            v2f bf;
            bf.x = w_att_e16[(k0 + koff)     * 16 + m16];
            bf.y = w_att_e16[(k0 + koff + 1) * 16 + m16];
            acc = __builtin_amdgcn_wmma_f32_16x16x4_f32(false, a, false, bf,
                                                        (short)0, acc, false, false);
        }
        const int h = m16;
        float sa = (h < HH) ? s_att[(size_t)bi * HH + h] : 0.f;
#pragma unroll
        for (int r = 0; r < 8; ++r) {
            int j = j0 + rbase + r;
            float v = 0.f;
            if (h < HH) {
                float sc = sa + d_att[((size_t)b * NN + j) * HH + h] + acc[r];
                sc = sc > 0.f ? sc : LRELU_SLOPE * sc;     // leaky relu
                v = mrow[j] ? sc : NEG_INF;
            }
            lds_att[j * 16 + h] = v;
        }
    }
    __syncthreads();

    // ---- Phase C: softmax over j, wave w handles head w ----------------
    {
        const int h = wave;                  // 0..7, all heads covered
        float vals[8];
        float mx = NEG_INF;
#pragma unroll
        for (int k = 0; k < 8; ++k) {
            vals[k] = lds_att[(lane + 32 * k) * 16 + h];
            mx = fmaxf(mx, vals[k]);
        }
#pragma unroll
        for (int off = 16; off > 0; off >>= 1)
            mx = fmaxf(mx, __shfl_xor(mx, off, 32));
        float sum = 0.f;
#pragma unroll
        for (int k = 0; k < 8; ++k) { vals[k] = __expf(vals[k] - mx); sum += vals[k]; }
#pragma unroll
        for (int off = 16; off > 0; off >>= 1)
            sum += __shfl_xor(sum, off, 32);
        float inv = 1.0f / sum;
#pragma unroll
        for (int k = 0; k < 8; ++k)
            lds_att[(lane + 32 * k) * 16 + h] = vals[k] * inv;
    }
    __syncthreads();

    // ---- Phase D / E (wave-uniform split) -----------------------------
    if (wave < 4) {
        // T[16x64] tile e0..e0+15 : att^T (A, from LDS) x edge_row (B, WGP$/L2 hit)
        const int e0 = wave << 4;
        v8f acc = {0.f,0.f,0.f,0.f,0.f,0.f,0.f,0.f};
        for (int k0 = 0; k0 < NN; k0 += 4) {
            v2f a, bf;
            a.x  = lds_att[(k0 + koff)     * 16 + m16];   // A[h=m16][j=k]
            a.y  = lds_att[(k0 + koff + 1) * 16 + m16];
            bf.x = erow[(size_t)(k0 + koff)     * EE + e0 + m16];
            bf.y = erow[(size_t)(k0 + koff + 1) * EE + e0 + m16];
            acc = __builtin_amdgcn_wmma_f32_16x16x4_f32(false, a, false, bf,
                                                        (short)0, acc, false, false);
        }
#pragma unroll
        for (int r = 0; r < 8; ++r)
            lds_T[(rbase + r) * EE + e0 + m16] = acc[r];
    } else {
        // agg_x: 128 lanes x 2 outputs each (x_proj is 1MB -> L2-resident)
        const int base = ((wave - 4) * 32 + lane) * 2;    // 0..254, same head for base,base+1
        const int h0 = base >> 5;
        const float* xp = x_proj + (size_t)b * NN * DD;
        float a0 = 0.f, a1 = 0.f;
        for (int j = 0; j < NN; ++j) {
            float w = lds_att[j * 16 + h0];
            v2f r = *(const v2f*)(xp + (size_t)j * DD + base);   // b64, 8B-aligned
            a0 += w * r.x;
            a1 += w * r.y;
        }
        lds_aggx[base]     = a0;
        lds_aggx[base + 1] = a1;
    }
    __syncthreads();

    // ---- Phase F: combine ---------------------------------------------
    {
        const int c = tid;              // 0..255 = h*32+d
        const int h = c >> 5;
        float acc = b_m[c] + bias_node[c] + lds_aggx[c];
#pragma unroll 8
        for (int e = 0; e < EE; ++e)
            acc += lds_T[h * EE + e] * W_m[(size_t)e * DD + c];
        x_out[(size_t)bi * DD + c] = acc;
    }
}

// ---------------------------------------------------------------------------
// out = LayerNorm(a + b) * g + beta   (one block per row of 256)
// ---------------------------------------------------------------------------
__global__ __launch_bounds__(256)
void add_ln_kernel(const float* __restrict__ a, const float* __restrict__ bsum,
                   const float* __restrict__ g, const float* __restrict__ beta,
                   float* __restrict__ out)
{
    __shared__ float red[256];
    const int row = blockIdx.x;
    const int t = threadIdx.x;
    float x = a[(size_t)row * DD + t] + bsum[(size_t)row * DD + t];
    red[t] = x; __syncthreads();
    for (int s = 128; s > 0; s >>= 1) { if (t < s) red[t] += red[t + s]; __syncthreads(); }
    float mean = red[0] * (1.0f / DD); __syncthreads();
    float xd = x - mean;
    red[t] = xd * xd; __syncthreads();
    for (int s = 128; s > 0; s >>= 1) { if (t < s) red[t] += red[t + s]; __syncthreads(); }
    float var = red[0] * (1.0f / DD);
    float rs = rsqrtf(var + 1e-5f);
    out[(size_t)row * DD + t] = xd * rs * g[t] + beta[t];
}

// ---------------------------------------------------------------------------
extern "C" void kernel_launch(void* const* d_in, const int* in_sizes, int n_in,
                              void* d_out, int out_size, void* d_ws, size_t ws_size,
                              hipStream_t stream)
{
    const float* node      = (const float*)d_in[0];
    const float* edge      = (const float*)d_in[1];
    const unsigned char* mask = (const unsigned char*)d_in[2];
    const float* W_v       = (const float*)d_in[3];
    const float* W_e       = (const float*)d_in[4];
    const float* W_m       = (const float*)d_in[5];
    const float* b_m       = (const float*)d_in[6];
    const float* att_src   = (const float*)d_in[7];
    const float* att_dst   = (const float*)d_in[8];
    const float* att_edge  = (const float*)d_in[9];
    const float* bias_node = (const float*)d_in[10];
    const float* W1        = (const float*)d_in[11];
    const float* b1        = (const float*)d_in[12];
    const float* W2        = (const float*)d_in[13];
    const float* b2        = (const float*)d_in[14];
    const float* g1        = (const float*)d_in[15];
    const float* be1       = (const float*)d_in[16];
    const float* g2        = (const float*)d_in[17];
    const float* be2       = (const float*)d_in[18];
    float* out = (float*)d_out;

    float* ws        = (float*)d_ws;
    float* x_proj    = ws;                    // 262144
    float* s_att     = x_proj + 262144;       // 8192
    float* d_att     = s_att + 8192;          // 8192
    float* w_att_e16 = d_att + 8192;          // 1024
    float* x_out     = w_att_e16 + 1024;      // 262144
    float* hbuf      = x_out + 262144;        // 262144
    float* ff1       = hbuf + 262144;         // 262144
    float* ff2       = ff1 + 262144;          // 262144

    const int M = BDIM * NN;                  // 1024
    const int gemm_blocks = (M / 16) * (DD / 16) / 8;   // 128
    dim3 blk(256);

    // 1) node projection
    gemm_f32_wmma<<<dim3(gemm_blocks), blk, 0, stream>>>(node, W_v, nullptr, x_proj, M, DD, DD, 0);
    // 2) per-node src/dst scores
    sdatt_kernel<<<dim3(32), blk, 0, stream>>>(x_proj, att_src, att_dst, s_att, d_att);
    // 3) fold W_e x att_edge
    fold_kernel<<<dim3(4), blk, 0, stream>>>(W_e, att_edge, w_att_e16);
    // 4) attention rows (streams edge_feats)
    attn_row_kernel<<<dim3(BDIM * NN), blk, 0, stream>>>(edge, mask, x_proj, s_att, d_att,
                                                         w_att_e16, W_m, b_m, bias_node, x_out);
    // 5) h = LN(node + x_out)
    add_ln_kernel<<<dim3(M), blk, 0, stream>>>(node, x_out, g1, be1, hbuf);
    // 6) ff1 = relu(h @ W1 + b1)
    gemm_f32_wmma<<<dim3(gemm_blocks), blk, 0, stream>>>(hbuf, W1, b1, ff1, M, DD, DD, 2);
    // 7) ff2 = ff1 @ W2 + b2
    gemm_f32_wmma<<<dim3(gemm_blocks), blk, 0, stream>>>(ff1, W2, b2, ff2, M, DD, DD, 1);
    // 8) out = LN(h + ff2)
    add_ln_kernel<<<dim3(M), blk, 0, stream>>>(hbuf, ff2, g2, be2, out);
}